// GCN_70300024701664
// MI455X (gfx1250) — compile-verified
//
#include <hip/hip_runtime.h>

typedef float v2f __attribute__((ext_vector_type(2)));
typedef float v8f __attribute__((ext_vector_type(8)));

#define HDIM 64
#define BN_EPS 1e-5f
#define LRELU_SLOPE 0.01f

// ---------------------------------------------------------------- zero
__global__ void zero_f32(float* __restrict__ p, long n) {
    long i = (long)blockIdx.x * blockDim.x + threadIdx.x;
    long stride = (long)gridDim.x * blockDim.x;
    for (; i < n; i += stride) p[i] = 0.0f;
}

// ---------------------------------------------------------------- scatter-mean accumulate
// One wave handles one edge per iteration: 32 lanes x float2 = 64 features.
// Gather x[src] (256B coalesced, L2-resident) and atomic-add into agg[dst].
__global__ void scatter_mean_accum(const float* __restrict__ x,
                                   const int* __restrict__ src,
                                   const int* __restrict__ dst,
                                   float* __restrict__ agg,
                                   float* __restrict__ cnt,
                                   int E) {
    int lane = threadIdx.x & 31;
    int wid  = (int)((blockIdx.x * blockDim.x + threadIdx.x) >> 5);
    int nw   = (int)((gridDim.x * blockDim.x) >> 5);
    for (int e = wid; e < E; e += nw) {
        int s = src[e];
        int d = dst[e];
        v2f v = *reinterpret_cast<const v2f*>(x + (long)s * HDIM + lane * 2);
        float* ap = agg + (long)d * HDIM + lane * 2;
        atomicAdd(ap + 0, v.x);
        atomicAdd(ap + 1, v.y);
        if (lane == 0) atomicAdd(cnt + d, 1.0f);
    }
}

// ---------------------------------------------------------------- fused SAGE GEMM
// out[m, :] = lrelu( (agg[m,:]/max(cnt[m],1)) @ Wl + bl + xin[m,:] @ Wr )
// One wave per 16-row tile; fp32 WMMA 16x16x4; weights transposed in LDS.
// A-frag layout (f32 16x4): lane L holds M = L&15, K = k0 + 2*(L>>4) + {0,1}
// C/D layout (f32 16x16):   vgpr r, lane L -> M = r + 8*(L>>4), N = L&15
__global__ void sage_gemm_lrelu(const float* __restrict__ agg,
                                const float* __restrict__ cnt,
                                const float* __restrict__ xin,
                                const float* __restrict__ Wl,
                                const float* __restrict__ bl,
                                const float* __restrict__ Wr,
                                float* __restrict__ out,
                                int N) {
    __shared__ __align__(16) float sWlT[HDIM * HDIM];  // transposed: [col][k]
    __shared__ __align__(16) float sWrT[HDIM * HDIM];
    for (int i = threadIdx.x; i < HDIM * HDIM; i += blockDim.x) {
        int r = i >> 6, c = i & 63;
        sWlT[c * HDIM + r] = Wl[i];
        sWrT[c * HDIM + r] = Wr[i];
    }
    __syncthreads();

    int lane = threadIdx.x & 31;
    int wave = threadIdx.x >> 5;
    int tile = blockIdx.x * (blockDim.x >> 5) + wave;
    int ntiles = N >> 4;            // N is a multiple of 16 (100000 = 6250*16)
    if (tile >= ntiles) return;     // wave-uniform guard (EXEC all-1s for WMMA)

    int m0  = tile << 4;
    int lm  = lane & 15;
    int g   = lane >> 4;
    int row = m0 + lm;

    float inv = 1.0f / fmaxf(cnt[row], 1.0f);

    v8f acc[4] = {};

    const float* arow = agg + (long)row * HDIM;
    const float* xrow = xin + (long)row * HDIM;

    // pass 1: (agg/cnt) @ Wl
    for (int k0 = 0; k0 < HDIM; k0 += 4) {
        int kk = k0 + 2 * g;
        v2f a = *reinterpret_cast<const v2f*>(arow + kk);
        a.x *= inv; a.y *= inv;
#pragma unroll
        for (int t = 0; t < 4; ++t) {
            int col = t * 16 + lm;
            v2f b = *reinterpret_cast<const v2f*>(&sWlT[col * HDIM + kk]);
            acc[t] = __builtin_amdgcn_wmma_f32_16x16x4_f32(
                false, a, false, b, (short)0, acc[t], false, false);
        }
    }
    // pass 2: += xin @ Wr
    for (int k0 = 0; k0 < HDIM; k0 += 4) {
        int kk = k0 + 2 * g;
        v2f a = *reinterpret_cast<const v2f*>(xrow + kk);
#pragma unroll
        for (int t = 0; t < 4; ++t) {
            int col = t * 16 + lm;
            v2f b = *reinterpret_cast<const v2f*>(&sWrT[col * HDIM + kk]);
            acc[t] = __builtin_amdgcn_wmma_f32_16x16x4_f32(
                false, a, false, b, (short)0, acc[t], false, false);
        }
    }

    // epilogue: bias + LeakyReLU, store
#pragma unroll
    for (int t = 0; t < 4; ++t) {
        int n = t * 16 + lm;
        float bias = bl[n];
#pragma unroll
        for (int r = 0; r < 8; ++r) {
            int m = m0 + r + 8 * g;
            float v = acc[t][r] + bias;
            v = (v > 0.0f) ? v : LRELU_SLOPE * v;
            out[(long)m * HDIM + n] = v;
        }
    }
}

// ---------------------------------------------------------------- BN batch stats
// stats[0..63] += sum over rows, stats[64..127] += sum of squares
__global__ void bn_stats(const float* __restrict__ h, float* __restrict__ stats, int N) {
    __shared__ float sS[256];
    __shared__ float sQ[256];
    int f  = threadIdx.x & 63;
    int rg = threadIdx.x >> 6;   // 0..3
    float s = 0.0f, q = 0.0f;
    for (int row = blockIdx.x * 4 + rg; row < N; row += gridDim.x * 4) {
        float v = h[(long)row * HDIM + f];
        s += v;
        q += v * v;
    }
    sS[threadIdx.x] = s;
    sQ[threadIdx.x] = q;
    __syncthreads();
    if (threadIdx.x < 64) {
        float ts = sS[f] + sS[f + 64] + sS[f + 128] + sS[f + 192];
        float tq = sQ[f] + sQ[f + 64] + sQ[f + 128] + sQ[f + 192];
        atomicAdd(&stats[f], ts);
        atomicAdd(&stats[f + 64], tq);
    }
}

// ---------------------------------------------------------------- BN apply (in place)
__global__ void bn_apply(float* __restrict__ h, const float* __restrict__ stats,
                         const float* __restrict__ gamma, const float* __restrict__ beta,
                         int N) {
    long i = (long)blockIdx.x * blockDim.x + threadIdx.x;
    long total = (long)N * HDIM;
    long stride = (long)gridDim.x * blockDim.x;
    float invN = 1.0f / (float)N;
    for (; i < total; i += stride) {
        int f = (int)(i & 63);
        float mu  = stats[f] * invN;
        float var = stats[f + 64] * invN - mu * mu;
        float rs  = rsqrtf(var + BN_EPS) * gamma[f];
        h[i] = (h[i] - mu) * rs + beta[f];
    }
}

// ---------------------------------------------------------------- final FC (H -> 1)
__global__ void fc_out(const float* __restrict__ h, const float* __restrict__ Wfc,
                       const float* __restrict__ bfc, float* __restrict__ out, int N) {
    int lane = threadIdx.x & 31;
    int row  = (int)((blockIdx.x * blockDim.x + threadIdx.x) >> 5);
    if (row >= N) return;
    v2f hv = *reinterpret_cast<const v2f*>(h + (long)row * HDIM + lane * 2);
    v2f wv = *reinterpret_cast<const v2f*>(Wfc + lane * 2);
    float p = hv.x * wv.x + hv.y * wv.y;
#pragma unroll
    for (int off = 16; off > 0; off >>= 1) p += __shfl_xor(p, off, 32);
    if (lane == 0) out[row] = p + bfc[0];
}

// ---------------------------------------------------------------- driver
extern "C" void kernel_launch(void* const* d_in, const int* in_sizes, int n_in,
                              void* d_out, int out_size, void* d_ws, size_t ws_size,
                              hipStream_t stream) {
    (void)n_in; (void)out_size; (void)ws_size;

    const int N = in_sizes[0] / HDIM;
    const int E = in_sizes[1] / 2;

    const float* x   = (const float*)d_in[0];
    const int*   ei  = (const int*)d_in[1];
    const int*   src = ei;
    const int*   dst = ei + E;
    const float* W0l = (const float*)d_in[2];
    const float* b0  = (const float*)d_in[3];
    const float* W0r = (const float*)d_in[4];
    const float* g0  = (const float*)d_in[5];
    const float* be0 = (const float*)d_in[6];
    const float* W1l = (const float*)d_in[7];
    const float* b1  = (const float*)d_in[8];
    const float* W1r = (const float*)d_in[9];
    const float* g1  = (const float*)d_in[10];
    const float* be1 = (const float*)d_in[11];
    const float* W2l = (const float*)d_in[12];
    const float* b2  = (const float*)d_in[13];
    const float* W2r = (const float*)d_in[14];
    const float* Wfc = (const float*)d_in[15];
    const float* bfc = (const float*)d_in[16];

    // workspace layout (floats): [agg N*64][cnt N][stats 128][hA N*64][hB N*64]
    float* agg   = (float*)d_ws;
    float* cnt   = agg + (size_t)N * HDIM;
    float* stats = cnt + N;
    float* hA    = stats + 128;
    float* hB    = hA + (size_t)N * HDIM;

    const long zcount = (long)N * HDIM + N + 128;   // agg + cnt + stats, contiguous
    const int  ZB = 1024, SB = 2048;
    const int  gemm_blocks = ((N >> 4) + 7) / 8;    // 8 waves/block, 1 tile/wave
    const int  bn_app_blocks = (int)(((long)N * HDIM + 255) / 256);
    const int  fc_blocks = (N * 32 + 255) / 256;

    // ---- layer 0: SAGE -> LeakyReLU -> BN
    zero_f32<<<ZB, 256, 0, stream>>>(agg, zcount);
    scatter_mean_accum<<<SB, 256, 0, stream>>>(x, src, dst, agg, cnt, E);
    sage_gemm_lrelu<<<gemm_blocks, 256, 0, stream>>>(agg, cnt, x, W0l, b0, W0r, hA, N);
    bn_stats<<<256, 256, 0, stream>>>(hA, stats, N);
    bn_apply<<<bn_app_blocks, 256, 0, stream>>>(hA, stats, g0, be0, N);

    // ---- layer 1: SAGE -> LeakyReLU -> BN
    zero_f32<<<ZB, 256, 0, stream>>>(agg, zcount);
    scatter_mean_accum<<<SB, 256, 0, stream>>>(hA, src, dst, agg, cnt, E);
    sage_gemm_lrelu<<<gemm_blocks, 256, 0, stream>>>(agg, cnt, hA, W1l, b1, W1r, hB, N);
    bn_stats<<<256, 256, 0, stream>>>(hB, stats, N);
    bn_apply<<<bn_app_blocks, 256, 0, stream>>>(hB, stats, g1, be1, N);

    // ---- layer 2: SAGE -> LeakyReLU (no BN)
    zero_f32<<<ZB, 256, 0, stream>>>(agg, zcount);
    scatter_mean_accum<<<SB, 256, 0, stream>>>(hB, src, dst, agg, cnt, E);
    sage_gemm_lrelu<<<gemm_blocks, 256, 0, stream>>>(agg, cnt, hB, W2l, b2, W2r, hA, N);

    // ---- final FC
    fc_out<<<fc_blocks, 256, 0, stream>>>(hA, Wfc, bfc, (float*)d_out, N);
}